// MultiHeadAttention_29824252903757
// MI455X (gfx1250) — compile-verified
//
#include <hip/hip_runtime.h>
#include <hip/hip_bf16.h>

// ---------------------------------------------------------------------------
// MHA forward for MI455X (gfx1250, wave32, WMMA).
// B=2, S=2048, D=1024, H=16, HD=64. fp32 end-to-end via V_WMMA_F32_16X16X4_F32.
// ---------------------------------------------------------------------------

typedef __attribute__((ext_vector_type(2))) float v2f;
typedef __attribute__((ext_vector_type(8))) float v8f;

// Only check on the DEVICE pass: __has_builtin() for amdgcn builtins is false
// during the host (x86) pass even though the call itself compiles there.
#if defined(__HIP_DEVICE_COMPILE__)
#if !__has_builtin(__builtin_amdgcn_wmma_f32_16x16x4_f32)
#error "wmma_f32_16x16x4_f32 builtin not available on this toolchain (device)"
#endif
#endif

#define WMMA_F32X4(a, b, c) \
    __builtin_amdgcn_wmma_f32_16x16x4_f32(false, (a), false, (b), (short)0, (c), false, false)

static constexpr int BB = 2;
static constexpr int SS = 2048;
static constexpr int DD = 1024;
static constexpr int HH = 16;
static constexpr int HD = 64;
static constexpr int MTOT = BB * SS;   // 4096 rows

// ---------------------------------------------------------------------------
// Kernel 1: fused Q/K/V projection.  Y[r,e] = sum_d X[r,d]*W[e,d] + b[e]
//   z==0 -> q  stored (B,H,S,HD)
//   z==1 -> kT stored (B,H,HD,S)   (transposed for coalesced QK^T B-frags)
//   z==2 -> v  stored (B,H,S,HD)
// 64x64 output tile per 128-thread (4-wave) block, K staged 32 at a time.
// ---------------------------------------------------------------------------
__global__ __launch_bounds__(128)
void qkv_proj_kernel(const float* __restrict__ Xq, const float* __restrict__ Xk,
                     const float* __restrict__ Xv,
                     const float* __restrict__ Wq, const float* __restrict__ Wk,
                     const float* __restrict__ Wv,
                     const float* __restrict__ bq, const float* __restrict__ bk,
                     const float* __restrict__ bv,
                     float* __restrict__ Oq, float* __restrict__ OkT,
                     float* __restrict__ Ov)
{
    __shared__ float Xs[64][33];   // [row][k]  (pad 33 -> conflict-free A frags)
    __shared__ float Ws[32][65];   // [k][n]    (pad 65 -> conflict-free B frags)

    const int z = blockIdx.z;
    const float* X    = (z == 0) ? Xq : (z == 1) ? Xk : Xv;
    const float* W    = (z == 0) ? Wq : (z == 1) ? Wk : Wv;
    const float* bias = (z == 0) ? bq : (z == 1) ? bk : bv;

    const int tileN = blockIdx.x * 64;
    const int tileM = blockIdx.y * 64;
    const int tid   = threadIdx.x;
    const int wave  = tid >> 5;
    const int lane  = tid & 31;
    const int m16   = lane & 15;           // A-row / B-col / C-col index
    const int kb    = (lane >> 4) * 2;     // K sub-offset within frag
    const int rowHalf = (lane >> 4) * 8;   // C/D row half select

    v8f acc[4];
    #pragma unroll
    for (int i = 0; i < 4; ++i) acc[i] = v8f{0.f,0.f,0.f,0.f,0.f,0.f,0.f,0.f};

    for (int kt = 0; kt < DD; kt += 32) {
        // stage X tile (64x32), coalesced
        for (int i = tid; i < 64 * 32; i += 128) {
            const int r = i >> 5, c = i & 31;
            Xs[r][c] = X[(size_t)(tileM + r) * DD + kt + c];
        }
        // stage W tile transposed: Ws[k][n] = W[n, k]
        for (int i = tid; i < 64 * 32; i += 128) {
            const int n = i >> 5, c = i & 31;
            Ws[c][n] = W[(size_t)(tileN + n) * DD + kt + c];
        }
        __syncthreads();

        #pragma unroll
        for (int k2 = 0; k2 < 32; k2 += 4) {
            v2f a;
            a.x = Xs[wave * 16 + m16][k2 + kb];
            a.y = Xs[wave * 16 + m16][k2 + kb + 1];
            #pragma unroll
            for (int nt = 0; nt < 4; ++nt) {
                v2f bf;
                bf.x = Ws[k2 + kb][nt * 16 + m16];
                bf.y = Ws[k2 + kb + 1][nt * 16 + m16];
                acc[nt] = WMMA_F32X4(a, bf, acc[nt]);
            }
        }
        __syncthreads();
    }

    // epilogue: bias + scatter into head layout
    #pragma unroll
    for (int nt = 0; nt < 4; ++nt) {
        #pragma unroll
        for (int j = 0; j < 8; ++j) {
            const int m  = j + rowHalf;
            const int gr = tileM + wave * 16 + m;     // row in [0, B*S)
            const int gc = tileN + nt * 16 + m16;     // col in [0, D)
            const float val = acc[nt][j] + bias[gc];
            const int bi = gr >> 11;                  // / S
            const int s  = gr & (SS - 1);
            const int hh = gc >> 6;                   // / HD
            const int hd = gc & (HD - 1);
            if (z == 1) {
                OkT[(((size_t)bi * HH + hh) * HD + hd) * SS + s] = val;
            } else {
                float* O = (z == 0) ? Oq : Ov;
                O[(((size_t)bi * HH + hh) * SS + s) * HD + hd] = val;
            }
        }
    }
}

// ---------------------------------------------------------------------------
// Kernel 2: fused attention per (b, h, 16-query tile).
// 256 threads = 8 waves. Score strip 16x2048 lives in LDS (stride 2052 pads
// away bank conflicts for the phase-3 A-matrix reads).
//   phase 1: S = (Q_tile @ K^T) * 0.125       (WMMA, waves split key range)
//   phase 2: mask + softmax (wave32 shfl reductions) + store attn weights
//   phase 3: ctx = P @ V                      (WMMA, waves split K dim)
// ---------------------------------------------------------------------------
static constexpr int SC_STRIDE = 2052;                    // 2052 % 64 == 4
static constexpr int QS_STRIDE = 68;                      // 68 % 64 == 4
static constexpr size_t ATTN_SMEM_FLOATS =
    (size_t)16 * SC_STRIDE + 16 * QS_STRIDE + 8 * 256;

__global__ __launch_bounds__(256)
void attention_kernel(const float* __restrict__ q, const float* __restrict__ kT,
                      const float* __restrict__ v,
                      const unsigned char* __restrict__ mask,
                      float* __restrict__ attn_w, float* __restrict__ ctx)
{
    extern __shared__ float smem[];
    float* sc   = smem;                          // [16][SC_STRIDE]
    float* qs   = smem + 16 * SC_STRIDE;         // [16][QS_STRIDE]
    float* part = qs + 16 * QS_STRIDE;           // [8][16][16]

    const int qt = blockIdx.x;                   // query tile (0..127)
    const int h  = blockIdx.y;
    const int b  = blockIdx.z;
    const int bh = b * HH + h;

    const float* qp = q  + ((size_t)bh * SS + (size_t)qt * 16) * HD;  // 16 x 64
    const float* kp = kT + (size_t)bh * HD * SS;                      // [hd][skey]
    const float* vp = v  + (size_t)bh * SS * HD;                      // [skey][hd]
    float* wout = attn_w + ((size_t)bh * SS + (size_t)qt * 16) * SS;  // 16 x 2048
    const unsigned char* mrow = mask + ((size_t)b * SS + (size_t)qt * 16) * SS;

    const int tid  = threadIdx.x;
    const int w    = tid >> 5;
    const int lane = tid & 31;
    const int n16  = lane & 15;
    const int kb   = (lane >> 4) * 2;
    const int rowHalf = (lane >> 4) * 8;

    // stage Q tile
    for (int i = tid; i < 16 * HD; i += 256)
        qs[(i >> 6) * QS_STRIDE + (i & 63)] = qp[i];
    __syncthreads();

    // ---- phase 1: scores; wave w covers key cols [w*256, w*256+256) ----
    for (int t = 0; t < 16; ++t) {
        const int n0 = w * 256 + t * 16;
        v8f acc = v8f{0.f,0.f,0.f,0.f,0.f,0.f,0.f,0.f};
        #pragma unroll
        for (int kk = 0; kk < HD; kk += 4) {
            v2f a;
            a.x = qs[n16 * QS_STRIDE + kk + kb];
            a.y = qs[n16 * QS_STRIDE + kk + kb + 1];
            v2f bf;
            bf.x = kp[(size_t)(kk + kb) * SS + n0 + n16];
            bf.y = kp[(size_t)(kk + kb + 1) * SS + n0 + n16];
            acc = WMMA_F32X4(a, bf, acc);
        }
        #pragma unroll
        for (int j = 0; j < 8; ++j)
            sc[(j + rowHalf) * SC_STRIDE + n0 + n16] = acc[j] * 0.125f;  // 1/sqrt(64)
    }
    __syncthreads();

    // ---- phase 2: mask + softmax over each 2048-row; 2 rows per wave ----
    for (int rr = w * 2; rr < w * 2 + 2; ++rr) {
        float mx = -3.4e38f;
        for (int c = lane; c < SS; c += 32) {
            float val = sc[rr * SC_STRIDE + c];
            if (mrow[(size_t)rr * SS + c]) { val = -1.0e9f; sc[rr * SC_STRIDE + c] = val; }
            mx = fmaxf(mx, val);
        }
        #pragma unroll
        for (int off = 16; off; off >>= 1) mx = fmaxf(mx, __shfl_xor(mx, off, 32));
        float sum = 0.f;
        for (int c = lane; c < SS; c += 32) sum += __expf(sc[rr * SC_STRIDE + c] - mx);
        #pragma unroll
        for (int off = 16; off; off >>= 1) sum += __shfl_xor(sum, off, 32);
        const float inv = 1.0f / sum;
        for (int c = lane; c < SS; c += 32) {
            const float wv = __expf(sc[rr * SC_STRIDE + c] - mx) * inv;
            sc[rr * SC_STRIDE + c] = wv;
            wout[(size_t)rr * SS + c] = wv;   // mandatory attn_weights output
        }
    }
    __syncthreads();

    // ---- phase 3: ctx(16x64) = P(16x2048) @ V(2048x64); waves split K ----
    {
        const int nt = w & 3;        // hd block of 16
        const int kh = w >> 2;       // key-dim half
        v8f acc = v8f{0.f,0.f,0.f,0.f,0.f,0.f,0.f,0.f};
        const int k0 = kh * 1024;
        for (int k = k0; k < k0 + 1024; k += 4) {
            v2f a;
            a.x = sc[n16 * SC_STRIDE + k + kb];
            a.y = sc[n16 * SC_STRIDE + k + kb + 1];
            v2f bf;
            bf.x = vp[(size_t)(k + kb) * HD + nt * 16 + n16];
            bf.y = vp[(size_t)(k + kb + 1) * HD + nt * 16 + n16];
            acc = WMMA_F32X4(a, bf, acc);
        }
        #pragma unroll
        for (int j = 0; j < 8; ++j)
            part[w * 256 + (j + rowHalf) * 16 + n16] = acc[j];
    }
    __syncthreads();

    // combine the two K halves, store ctx in (B, S, D) layout (head merge)
    for (int e = tid; e < 16 * HD; e += 256) {
        const int m = e >> 6, c = e & 63;
        const int nt2 = c >> 4, nn = c & 15;
        const float val = part[nt2 * 256 + m * 16 + nn]
                        + part[(nt2 + 4) * 256 + m * 16 + nn];
        ctx[((size_t)b * SS + (size_t)qt * 16 + m) * DD + (size_t)h * HD + c] = val;
    }
}

// ---------------------------------------------------------------------------
// Kernel 3: output projection. out[r,e] = sum_d ctx[r,d]*Wo[e,d] + bo[e]
// Same WMMA tiling as kernel 1, plain row-major store.
// ---------------------------------------------------------------------------
__global__ __launch_bounds__(128)
void out_proj_kernel(const float* __restrict__ X, const float* __restrict__ W,
                     const float* __restrict__ bias, float* __restrict__ O)
{
    __shared__ float Xs[64][33];
    __shared__ float Ws[32][65];

    const int tileN = blockIdx.x * 64;
    const int tileM = blockIdx.y * 64;
    const int tid   = threadIdx.x;
    const int wave  = tid >> 5;
    const int lane  = tid & 31;
    const int m16   = lane & 15;
    const int kb    = (lane >> 4) * 2;
    const int rowHalf = (lane >> 4) * 8;

    v8f acc[4];
    #pragma unroll
    for (int i = 0; i < 4; ++i) acc[i] = v8f{0.f,0.f,0.f,0.f,0.f,0.f,0.f,0.f};

    for (int kt = 0; kt < DD; kt += 32) {
        for (int i = tid; i < 64 * 32; i += 128) {
            const int r = i >> 5, c = i & 31;
            Xs[r][c] = X[(size_t)(tileM + r) * DD + kt + c];
        }
        for (int i = tid; i < 64 * 32; i += 128) {
            const int n = i >> 5, c = i & 31;
            Ws[c][n] = W[(size_t)(tileN + n) * DD + kt + c];
        }
        __syncthreads();

        #pragma unroll
        for (int k2 = 0; k2 < 32; k2 += 4) {
            v2f a;
            a.x = Xs[wave * 16 + m16][k2 + kb];
            a.y = Xs[wave * 16 + m16][k2 + kb + 1];
            #pragma unroll
            for (int nt = 0; nt < 4; ++nt) {
                v2f bf;
                bf.x = Ws[k2 + kb][nt * 16 + m16];
                bf.y = Ws[k2 + kb + 1][nt * 16 + m16];
                acc[nt] = WMMA_F32X4(a, bf, acc[nt]);
            }
        }
        __syncthreads();
    }

    #pragma unroll
    for (int nt = 0; nt < 4; ++nt) {
        #pragma unroll
        for (int j = 0; j < 8; ++j) {
            const int m  = j + rowHalf;
            const int gr = tileM + wave * 16 + m;
            const int gc = tileN + nt * 16 + m16;
            O[(size_t)gr * DD + gc] = acc[nt][j] + bias[gc];
        }
    }
}

// ---------------------------------------------------------------------------
// Host side
// ---------------------------------------------------------------------------
extern "C" void kernel_launch(void* const* d_in, const int* in_sizes, int n_in,
                              void* d_out, int out_size, void* d_ws, size_t ws_size,
                              hipStream_t stream)
{
    const float* Q  = (const float*)d_in[0];
    const float* K  = (const float*)d_in[1];
    const float* V  = (const float*)d_in[2];
    const unsigned char* mask = (const unsigned char*)d_in[3];
    const float* wq = (const float*)d_in[4];
    const float* bq = (const float*)d_in[5];
    const float* wk = (const float*)d_in[6];
    const float* bk = (const float*)d_in[7];
    const float* wv = (const float*)d_in[8];
    const float* bv = (const float*)d_in[9];
    const float* wo = (const float*)d_in[10];
    const float* bo = (const float*)d_in[11];

    float* out    = (float*)d_out;                              // B*S*D
    float* attn_w = out + (size_t)BB * SS * DD;                 // B*H*S*S

    const size_t PLANE = (size_t)BB * SS * DD;                  // 4M floats
    float* q_ws   = (float*)d_ws;
    float* kT_ws  = q_ws  + PLANE;
    float* v_ws   = kT_ws + PLANE;
    float* ctx_ws = v_ws  + PLANE;                              // 64 MB total

    // 1) fused QKV projections (z selects q / kT / v)
    qkv_proj_kernel<<<dim3(DD / 64, MTOT / 64, 3), 128, 0, stream>>>(
        Q, K, V, wq, wk, wv, bq, bk, bv, q_ws, kT_ws, v_ws);

    // 2) fused attention (scores + softmax + weights store + P@V)
    const size_t smem_bytes = ATTN_SMEM_FLOATS * sizeof(float); // ~140.5 KB LDS
    attention_kernel<<<dim3(SS / 16, HH, BB), 256, smem_bytes, stream>>>(
        q_ws, kT_ws, v_ws, mask, attn_w, ctx_ws);

    // 3) output projection
    out_proj_kernel<<<dim3(DD / 64, MTOT / 64, 1), 128, 0, stream>>>(
        ctx_ws, wo, bo, out);
}